// WindowMHSA_1872605741619
// MI455X (gfx1250) — compile-verified
//
#include <hip/hip_runtime.h>
#include <hip/hip_bf16.h>

typedef __attribute__((ext_vector_type(2))) float v2f;
typedef __attribute__((ext_vector_type(4))) float v4f;
typedef __attribute__((ext_vector_type(8))) float v8f;

namespace {
constexpr int kWS = 7;
constexpr int kNH = 4;
constexpr int kC  = 128;
constexpr int kL  = 49;   // tokens per window
constexpr int kLP = 64;   // padded to multiple of 16
constexpr int kHD = 32;   // head dim

// LDS strides in floats; stride % 64 == 4 -> conflict-free frag access
constexpr int SA_XSTRIDE = 132;                      // X view / Q view
constexpr int QOFF       = kLP * SA_XSTRIDE;         // 8448: Q lives after X
constexpr int SA_PSTRIDE = 68;                       // P (softmax probs) view
constexpr int SA_SIZE    = kNH * kLP * SA_PSTRIDE;   // 17408 (>= QOFF + 8448)
constexpr int SKV_STRIDE = 260;                      // K at cols 0..127, V at 128..255
constexpr int SKV_SIZE   = kLP * SKV_STRIDE;         // 16640
constexpr int SO_STRIDE  = 132;                      // O view inside sKV
constexpr int SB_SIZE    = (2 * kWS - 1) * (2 * kWS - 1) * kNH; // 676

static_assert(SA_SIZE >= QOFF + kLP * SA_XSTRIDE, "Q must fit behind X");
static_assert(kLP * SO_STRIDE <= SKV_SIZE, "O must fit in KV buffer");
}

// D = A(16x4 f32) * B(4x16 f32) + C(16x16 f32), full-precision matrix core
__device__ __forceinline__ v8f wmma4(v2f a, v2f b, v8f c) {
  return __builtin_amdgcn_wmma_f32_16x16x4_f32(false, a, false, b, (short)0, c,
                                               false, false);
}

__global__ __launch_bounds__(256) void swin_mhsa_fused(
    const float* __restrict__ x, const float* __restrict__ w_qkv,
    const float* __restrict__ b_qkv, const float* __restrict__ bias_table,
    const float* __restrict__ w_proj, const float* __restrict__ b_proj,
    float* __restrict__ out) {
  __shared__ float sA[SA_SIZE];    // X (0..8447) + Q (8448..16895), later P
  __shared__ float sKV[SKV_SIZE];  // K|V, later attention output O
  __shared__ float sBias[SB_SIZE];

  const int tid  = threadIdx.x;
  const int lane = tid & 31;
  const int wid  = tid >> 5;  // 0..7
  const int l16  = lane & 15;
  const int hi   = lane >> 4;  // half-wave id

  const int win = blockIdx.x;
  const int b   = win >> 8;
  const int wh  = (win >> 4) & 15;
  const int ww  = win & 15;
  const long gbase = (((long)b * 112 + (long)wh * kWS) * 112 + (long)ww * kWS) * kC;

  // ---------------- Phase 0: stage X window (rows 49..63 zero) + bias table
  {
    const int row0 = tid >> 5;         // 0..7
    const int col  = (tid & 31) * 4;   // 0..124
#pragma unroll
    for (int it = 0; it < 8; ++it) {
      const int l = row0 + it * 8;     // 0..63
      v4f v = {0.f, 0.f, 0.f, 0.f};
      if (l < kL) {
        const int r = l / kWS, cc = l % kWS;
        v = *(const v4f*)(x + gbase + ((long)r * 112 + cc) * kC + col);
      }
      *(v4f*)&sA[l * SA_XSTRIDE + col] = v;
    }
    for (int i = tid; i < SB_SIZE; i += 256) sBias[i] = bias_table[i];
  }
  __syncthreads();

  // ---------------- Phase 1: QKV = X @ w_qkv + b_qkv  (64x128 @ 128x384)
  // wave 'wid' owns 16-column strip wid*16 of Q (t=0), K (t=1), V (t=2)
#pragma unroll
  for (int t = 0; t < 3; ++t) {
    const int n0  = wid * 16;
    const int wqn = t * 128 + n0 + l16;  // column in w_qkv
    v2f bf[32];
#pragma unroll
    for (int ks = 0; ks < 32; ++ks) {
      const int k = 4 * ks + 2 * hi;
      bf[ks].x = w_qkv[k * 384 + wqn];
      bf[ks].y = w_qkv[(k + 1) * 384 + wqn];
    }
    const float bias = b_qkv[wqn];
#pragma unroll
    for (int mt = 0; mt < 4; ++mt) {
      v8f acc = {};
#pragma unroll
      for (int ks = 0; ks < 32; ++ks) {
        const v2f a =
            *(const v2f*)&sA[(mt * 16 + l16) * SA_XSTRIDE + 4 * ks + 2 * hi];
        acc = wmma4(a, bf[ks], acc);
      }
#pragma unroll
      for (int r = 0; r < 8; ++r) {
        const int row = mt * 16 + r + 8 * hi;
        const float v = acc[r] + bias;
        if (t == 0)
          sA[QOFF + row * SA_XSTRIDE + n0 + l16] = v;           // Q
        else if (t == 1)
          sKV[row * SKV_STRIDE + n0 + l16] = v;                 // K
        else
          sKV[row * SKV_STRIDE + 128 + n0 + l16] = v;           // V
      }
    }
  }
  __syncthreads();

  // ---------------- Phase 2a: S = scale*Q@K^T + bias, softmax (2 waves/head)
  const int h    = wid >> 1;
  const int half = wid & 1;
  const int m0b  = 32 * half;
  const float scale = 0.17677669529663687f;  // 1/sqrt(32)

  float P[2][4][8];
#pragma unroll
  for (int mt2 = 0; mt2 < 2; ++mt2) {
    v2f aq[8];
#pragma unroll
    for (int ks = 0; ks < 8; ++ks)
      aq[ks] = *(const v2f*)&sA[QOFF + (m0b + mt2 * 16 + l16) * SA_XSTRIDE +
                                h * kHD + 4 * ks + 2 * hi];
    v8f Sacc[4];
#pragma unroll
    for (int nt = 0; nt < 4; ++nt) {
      v8f acc = {};
#pragma unroll
      for (int ks = 0; ks < 8; ++ks) {
        // B = K^T: contiguous pair along K's feature dim
        const v2f bk = *(const v2f*)&sKV[(nt * 16 + l16) * SKV_STRIDE +
                                         h * kHD + 4 * ks + 2 * hi];
        acc = wmma4(aq[ks], bk, acc);
      }
      Sacc[nt] = acc;
    }
#pragma unroll
    for (int r = 0; r < 8; ++r) {
      const int i  = m0b + mt2 * 16 + r + 8 * hi;  // query token
      const int ii = i < kL ? i : (kL - 1);
      const int ri = ii / kWS, ci = ii % kWS;
      float rv[4];
#pragma unroll
      for (int nt = 0; nt < 4; ++nt) {
        const int j = nt * 16 + l16;  // key token
        float v = Sacc[nt][r] * scale;
        if (j < kL) {
          const int rj = j / kWS, cj = j % kWS;
          const int idx =
              (ri - rj + kWS - 1) * (2 * kWS - 1) + (ci - cj + kWS - 1);
          v += sBias[idx * kNH + h];
        } else {
          v = -1.0e30f;  // mask padded keys
        }
        rv[nt] = v;
      }
      float m = fmaxf(fmaxf(rv[0], rv[1]), fmaxf(rv[2], rv[3]));
      m = fmaxf(m, __shfl_xor(m, 1, 32));
      m = fmaxf(m, __shfl_xor(m, 2, 32));
      m = fmaxf(m, __shfl_xor(m, 4, 32));
      m = fmaxf(m, __shfl_xor(m, 8, 32));
      float s = 0.f;
#pragma unroll
      for (int nt = 0; nt < 4; ++nt) {
        rv[nt] = __expf(rv[nt] - m);
        s += rv[nt];
      }
      s += __shfl_xor(s, 1, 32);
      s += __shfl_xor(s, 2, 32);
      s += __shfl_xor(s, 4, 32);
      s += __shfl_xor(s, 8, 32);
      const float inv = 1.0f / s;
#pragma unroll
      for (int nt = 0; nt < 4; ++nt) P[mt2][nt][r] = rv[nt] * inv;
    }
  }
  __syncthreads();  // all Q/K reads done -> sA reusable for P
#pragma unroll
  for (int mt2 = 0; mt2 < 2; ++mt2)
#pragma unroll
    for (int nt = 0; nt < 4; ++nt)
#pragma unroll
      for (int r = 0; r < 8; ++r)
        sA[(h * kLP + m0b + mt2 * 16 + r + 8 * hi) * SA_PSTRIDE + nt * 16 +
           l16] = P[mt2][nt][r];
  __syncthreads();

  // ---------------- Phase 2b: O = P @ V  (64x64 @ 64x32 per head)
  v8f O[2][2];
  {
    v2f aP[2][16];
#pragma unroll
    for (int mt2 = 0; mt2 < 2; ++mt2)
#pragma unroll
      for (int ks = 0; ks < 16; ++ks)
        aP[mt2][ks] =
            *(const v2f*)&sA[(h * kLP + m0b + mt2 * 16 + l16) * SA_PSTRIDE +
                             4 * ks + 2 * hi];
#pragma unroll
    for (int ot = 0; ot < 2; ++ot) {
      v2f bv[16];
#pragma unroll
      for (int ks = 0; ks < 16; ++ks) {
        const int k0 = 4 * ks + 2 * hi;
        bv[ks].x = sKV[k0 * SKV_STRIDE + 128 + h * kHD + ot * 16 + l16];
        bv[ks].y = sKV[(k0 + 1) * SKV_STRIDE + 128 + h * kHD + ot * 16 + l16];
      }
#pragma unroll
      for (int mt2 = 0; mt2 < 2; ++mt2) {
        v8f acc = {};
#pragma unroll
        for (int ks = 0; ks < 16; ++ks) acc = wmma4(aP[mt2][ks], bv[ks], acc);
        O[mt2][ot] = acc;
      }
    }
  }
  __syncthreads();  // all V reads done -> sKV reusable for O
#pragma unroll
  for (int mt2 = 0; mt2 < 2; ++mt2)
#pragma unroll
    for (int ot = 0; ot < 2; ++ot)
#pragma unroll
      for (int r = 0; r < 8; ++r)
        sKV[(m0b + mt2 * 16 + r + 8 * hi) * SO_STRIDE + h * kHD + ot * 16 +
            l16] = O[mt2][ot][r];
  __syncthreads();

  // ---------------- Phase 3: out = O @ w_proj + b_proj, window-merge store
  {
    const int n0 = wid * 16;
    v2f bf[32];
#pragma unroll
    for (int ks = 0; ks < 32; ++ks) {
      const int k = 4 * ks + 2 * hi;
      bf[ks].x = w_proj[k * kC + n0 + l16];
      bf[ks].y = w_proj[(k + 1) * kC + n0 + l16];
    }
    const float bias = b_proj[n0 + l16];
#pragma unroll
    for (int mt = 0; mt < 4; ++mt) {
      v8f acc = {};
#pragma unroll
      for (int ks = 0; ks < 32; ++ks) {
        const v2f a =
            *(const v2f*)&sKV[(mt * 16 + l16) * SO_STRIDE + 4 * ks + 2 * hi];
        acc = wmma4(a, bf[ks], acc);
      }
#pragma unroll
      for (int r = 0; r < 8; ++r) {
        const int l = mt * 16 + r + 8 * hi;
        if (l < kL) {
          const int rr = l / kWS, cc = l % kWS;
          out[gbase + ((long)rr * 112 + cc) * kC + n0 + l16] = acc[r] + bias;
        }
      }
    }
  }
}

extern "C" void kernel_launch(void* const* d_in, const int* in_sizes, int n_in,
                              void* d_out, int out_size, void* d_ws,
                              size_t ws_size, hipStream_t stream) {
  (void)n_in; (void)out_size; (void)d_ws; (void)ws_size;
  const float* x          = (const float*)d_in[0];
  const float* w_qkv      = (const float*)d_in[1];
  const float* b_qkv      = (const float*)d_in[2];
  const float* bias_table = (const float*)d_in[3];
  const float* w_proj     = (const float*)d_in[4];
  const float* b_proj     = (const float*)d_in[5];
  float* out              = (float*)d_out;

  const int batch    = in_sizes[0] / (112 * 112 * 128);  // 32
  const int nWindows = batch * 16 * 16;                  // 8192
  hipLaunchKernelGGL(swin_mhsa_fused, dim3(nWindows), dim3(256), 0, stream, x,
                     w_qkv, b_qkv, bias_table, w_proj, b_proj, out);
}